// SimpleRNN_85710367359479
// MI455X (gfx1250) — compile-verified
//
#include <hip/hip_runtime.h>

typedef __attribute__((ext_vector_type(2))) float v2f;
typedef __attribute__((ext_vector_type(8))) float v8f;
typedef __attribute__((ext_vector_type(4))) unsigned int u4v;
typedef __attribute__((ext_vector_type(4))) int i4v;
typedef __attribute__((ext_vector_type(8))) int i8v;

#define D 2048
#define SEQ 8192
#define BLK 128   // timesteps per sequence block
#define NB 64     // number of sequence blocks (SEQ / BLK)

// ---------------------------------------------------------------------------
// TDM 2D tile load: global (row-major, stride in elements) -> LDS.
// Builds D# group0/group1 per CDNA5 ISA.
// data_size = 4B floats. Optional LDS pad: +1 dword every 16 dwords
// (pad_interval=3 -> 16 dwords, pad_amount=0 -> 1 dword) so a [rows][16]
// tile lands as [rows][17] (bank-conflict-free fragment reads, 17 ⟂ 64).
// Tracked with TENSORcnt; in-order per wave.
// ---------------------------------------------------------------------------
__device__ __forceinline__ void tdm_load_tile_2d(
    unsigned lds_byte_off, const float* gptr,
    unsigned tile_d0, unsigned tile_d1,
    unsigned long long stride_elems, bool pad)
{
    const unsigned long long ga = (unsigned long long)gptr;
    const unsigned td0 = (unsigned)stride_elems;  // tensor_dim0 (>= tile extent)
    const unsigned td1 = 0xFFFFFu;                // tensor_dim1 (always in-bounds)

    u4v g0;
    g0[0] = 1u;                                   // count=1, user mode, no gather
    g0[1] = lds_byte_off;                         // lds_addr
    g0[2] = (unsigned)(ga & 0xFFFFFFFFu);         // global_addr[31:0]
    g0[3] = (unsigned)((ga >> 32) & 0x1FFFFFFu)   // global_addr[56:32]
          | (2u << 30);                           // type = 2 ("image")

    i8v g1;
    unsigned d0 = (2u << 16);                     // data_size = 4 bytes
    if (pad) d0 |= (1u << 20) | (3u << 22);       // pad_enable, every 16 dwords, +1 dword
    g1[0] = (int)d0;                              // wg_mask=0, no atomic barrier
    g1[1] = (int)((td0 & 0xFFFFu) << 16);         // abar_addr=0 | tensor_dim0 lo
    g1[2] = (int)((td0 >> 16) | ((td1 & 0xFFFFu) << 16));
    g1[3] = (int)((td1 >> 16) | (tile_d0 << 16)); // tensor_dim1 hi | tile_dim0
    g1[4] = (int)(tile_d1 & 0xFFFFu);             // tile_dim1 | tile_dim2=0
    g1[5] = (int)(unsigned)(stride_elems & 0xFFFFFFFFu);   // dim0_stride lo
    g1[6] = (int)(unsigned)((stride_elems >> 32) & 0xFFFFu); // dim0_stride hi | dim1_stride=0
    g1[7] = 0;

    i4v z4 = {0, 0, 0, 0};
#if __clang_major__ >= 23
    i8v z8 = {0, 0, 0, 0, 0, 0, 0, 0};
    __builtin_amdgcn_tensor_load_to_lds(g0, g1, z4, z4, z8, 0);
#else
    __builtin_amdgcn_tensor_load_to_lds(g0, g1, z4, z4, 0);
#endif
}

// ---------------------------------------------------------------------------
// TDM-fed, LDS-tiled, double-buffered fp32 WMMA GEMM.
//   C[m,n] = sum_k A[m,k] * opB[k,n]  (+ optional RNN epilogue)
//   TRANSB = true : opB[k,n] = B[n*ldb + k]
//   TRANSB = false: opB[k,n] = B[k*ldb + n]
//   RNN_EPI: C[m,n] += Ug[(n*BLK + step)*D + m]
// Block = 256 threads = 8 waves; block tile 128x64, K tile 16.
// Each wave computes a 16x64 strip (4 accumulators, A-frag reuse x4).
// Wave 0 drives the TDM pipeline: issue tile kt+1, s_wait_tensorcnt<=2 for
// tile kt (in-order), barrier, 16 WMMAs from LDS, barrier.
// ---------------------------------------------------------------------------
template <bool TRANSB, bool RNN_EPI>
__global__ __launch_bounds__(256) void gemm_tdm(
    const float* __restrict__ A, int lda,
    const float* __restrict__ B, int ldb,
    float* __restrict__ C, int ldc, int K,
    const float* __restrict__ Ug, int step)
{
    constexpr int BM = 128, BN = 64, BK = 16;
    __shared__ float As[2][BM][BK + 1];                       // TDM-padded stride 17
    __shared__ float Bs[2][TRANSB ? BN : BK][TRANSB ? (BK + 1) : BN];

    const int tid  = threadIdx.x;
    const int wave = tid >> 5;
    const int lane = tid & 31;
    const int hh   = lane >> 4;        // lane-half selects K pair
    const int lc   = lane & 15;
    const int mblk = blockIdx.x * BM;
    const int n0   = blockIdx.y * BN;
    const int ml   = wave * 16 + lc;   // row within LDS A tile

    v8f acc[4] = {};

    const int NT = K / BK;

    if (tid < 32) {  // wave 0 primes the pipeline
        tdm_load_tile_2d((unsigned)(unsigned long long)(const void*)&As[0][0][0],
                         A + (size_t)mblk * lda, BK, BM,
                         (unsigned long long)lda, true);
        if (TRANSB)
            tdm_load_tile_2d((unsigned)(unsigned long long)(const void*)&Bs[0][0][0],
                             B + (size_t)n0 * ldb, BK, BN,
                             (unsigned long long)ldb, true);
        else
            tdm_load_tile_2d((unsigned)(unsigned long long)(const void*)&Bs[0][0][0],
                             B + n0, BN, BK,
                             (unsigned long long)ldb, false);
    }

    for (int kt = 0; kt < NT; ++kt) {
        const int cb = kt & 1;
        if (tid < 32) {
            if (kt + 1 < NT) {                     // issue next tile, then wait current
                const int nb = cb ^ 1;
                const int kk = (kt + 1) * BK;
                tdm_load_tile_2d((unsigned)(unsigned long long)(const void*)&As[nb][0][0],
                                 A + (size_t)mblk * lda + kk, BK, BM,
                                 (unsigned long long)lda, true);
                if (TRANSB)
                    tdm_load_tile_2d((unsigned)(unsigned long long)(const void*)&Bs[nb][0][0],
                                     B + (size_t)n0 * ldb + kk, BK, BN,
                                     (unsigned long long)ldb, true);
                else
                    tdm_load_tile_2d((unsigned)(unsigned long long)(const void*)&Bs[nb][0][0],
                                     B + (size_t)kk * ldb + n0, BN, BK,
                                     (unsigned long long)ldb, false);
                __builtin_amdgcn_s_wait_tensorcnt(2);   // tile kt complete (in-order)
            } else {
                __builtin_amdgcn_s_wait_tensorcnt(0);
            }
        }
        __syncthreads();    // buffer cb ready for all waves

#pragma unroll
        for (int ks = 0; ks < 4; ++ks) {
            const int kl = ks * 4 + 2 * hh;
            v2f a;
            a.x = As[cb][ml][kl];
            a.y = As[cb][ml][kl + 1];
#pragma unroll
            for (int nt = 0; nt < 4; ++nt) {
                const int nn = nt * 16 + lc;
                v2f b;
                if constexpr (TRANSB) {
                    b.x = Bs[cb][nn][kl];
                    b.y = Bs[cb][nn][kl + 1];
                } else {
                    b.x = Bs[cb][kl][nn];
                    b.y = Bs[cb][kl + 1][nn];
                }
                acc[nt] = __builtin_amdgcn_wmma_f32_16x16x4_f32(
                    false, a, false, b, (short)0, acc[nt], false, false);
            }
        }
        __syncthreads();    // done reading buffer cb before TDM reuses it
    }

#pragma unroll
    for (int nt = 0; nt < 4; ++nt) {
        const int n = n0 + nt * 16 + lc;
#pragma unroll
        for (int v = 0; v < 8; ++v) {
            const int m = mblk + wave * 16 + v + 8 * hh;
            float val = acc[nt][v];
            if (RNN_EPI)
                val += Ug[((size_t)n * BLK + step) * D + m];
            C[(size_t)m * ldc + n] = val;
        }
    }
}

// ---------------------------------------------------------------------------
// Carry step:  y[d] = dot(M[d,:], x) + V[d*NB + j]      (M is 16.8 MB, L2-hot)
// ---------------------------------------------------------------------------
__global__ __launch_bounds__(256) void carry_matvec(
    const float* __restrict__ Mm, const float* __restrict__ x,
    const float* __restrict__ V, int j, float* __restrict__ y)
{
    const int d = blockIdx.x * 256 + threadIdx.x;
    const float4* row = reinterpret_cast<const float4*>(Mm + (size_t)d * D);
    const float4* xv  = reinterpret_cast<const float4*>(x);
    float s = 0.f;
#pragma unroll 4
    for (int k = 0; k < D / 4; ++k) {
        float4 r = row[k], xx = xv[k];
        s += r.x * xx.x + r.y * xx.y + r.z * xx.z + r.w * xx.w;
    }
    y[d] = s + V[(size_t)d * NB + j];
}

// ---------------------------------------------------------------------------
// h_final = sum_r Wh^(8191-r) u_r, computed as:
//   U = X @ Wx^T                 (1 GEMM)
//   M = Wh^128                   (7 squaring GEMMs)
//   V <- Wh*V + U_i, i=0..127    (64 blocks batched as V columns; stream-ordered)
//   c <- M*c + v_j,  j=0..63     (final carry; last write -> d_out)
// ---------------------------------------------------------------------------
extern "C" void kernel_launch(void* const* d_in, const int* in_sizes, int n_in,
                              void* d_out, int out_size, void* d_ws, size_t ws_size,
                              hipStream_t stream)
{
    const float* X  = (const float*)d_in[0];  // [8192, 2048]
    const float* Wx = (const float*)d_in[1];  // [2048, 2048]
    const float* Wh = (const float*)d_in[2];  // [2048, 2048]
    float* out = (float*)d_out;               // [2048]

    float* U  = (float*)d_ws;                       // SEQ*D
    float* P0 = U + (size_t)SEQ * D;                // D*D
    float* P1 = P0 + (size_t)D * D;                 // D*D
    float* V0 = P1 + (size_t)D * D;                 // D*NB
    float* V1 = V0 + (size_t)D * NB;                // D*NB
    float* c0 = V1 + (size_t)D * NB;                // D
    float* c1 = c0 + D;                             // D

    // Phase 1: U = X @ Wx^T
    {
        dim3 grid(SEQ / 128, D / 64);
        gemm_tdm<true, false><<<grid, 256, 0, stream>>>(X, D, Wx, D, U, D, D,
                                                        nullptr, 0);
    }

    // Phase 2: M = Wh^128 via 7 repeated squarings (ping-pong P0/P1)
    dim3 gsq(D / 128, D / 64);
    gemm_tdm<false, false><<<gsq, 256, 0, stream>>>(Wh, D, Wh, D, P0, D, D,
                                                    nullptr, 0);      // Wh^2
    const float* Msrc = P0;
    float* Mdst = P1;
    for (int s = 1; s < 7; ++s) {                                     // ^4..^128
        gemm_tdm<false, false><<<gsq, 256, 0, stream>>>(Msrc, D, Msrc, D,
                                                        Mdst, D, D, nullptr, 0);
        const float* t = Msrc; Msrc = Mdst; Mdst = (float*)t;
    }
    // Msrc -> Wh^128

    // Phase 3: batched intra-block recurrence  V <- Wh*V + U_i, i = 0..127
    hipMemsetAsync(V0, 0, (size_t)D * NB * sizeof(float), stream);
    {
        dim3 grid(D / 128, 1);
        float* vb[2] = {V0, V1};
        for (int i = 0; i < BLK; ++i) {
            gemm_tdm<false, true><<<grid, 256, 0, stream>>>(
                Wh, D, vb[i & 1], NB, vb[(i + 1) & 1], NB, D, U, i);
        }
        // BLK even -> final V in V0
    }

    // Phase 4: inter-block carry  c <- M*c + v_j, j = 0..63 (last -> out)
    hipMemsetAsync(c0, 0, (size_t)D * sizeof(float), stream);
    {
        float* cb[2] = {c0, c1};
        for (int j = 0; j < NB; ++j) {
            float* y = (j == NB - 1) ? out : cb[(j + 1) & 1];
            carry_matvec<<<D / 256, 256, 0, stream>>>(Msrc, cb[j & 1], V0, j, y);
        }
    }
}